// LazyCardiacDVFWarpModule_56495999812286
// MI455X (gfx1250) — compile-verified
//
#include <hip/hip_runtime.h>

// Problem constants (reference: D=H=W=128, NUM_STEPS=5)
#define DIM   128
#define NVOX  (DIM * DIM * DIM)          // 2,097,152 voxels
#define NUM_STEPS 5
#define VPT   4                          // voxels per thread (b128 vectorization)

// ---------------------------------------------------------------------------
// Field layout for ddf buffers: interleaved vec3  ws[3*idx + c],
// idx = (z<<14) + (y<<7) + x, c in {dz, dy, dx}.
// Sampling semantics match grid_sample(padding_mode='zeros',
// align_corners=True) in voxel space: per-corner bounds mask folded into the
// trilinear weight, indices clamped so loads are always in-bounds.
// ---------------------------------------------------------------------------

__device__ __forceinline__ void axis_corners(float c, int& i0c, int& i1c,
                                             float& w0, float& w1) {
    float f = floorf(c);
    float t = c - f;
    int i0 = (int)f;
    int i1 = i0 + 1;
    float m0 = (i0 >= 0 && i0 < DIM) ? 1.0f : 0.0f;
    float m1 = (i1 >= 0 && i1 < DIM) ? 1.0f : 0.0f;
    w0 = (1.0f - t) * m0;
    w1 = t * m1;
    i0c = min(max(i0, 0), DIM - 1);
    i1c = min(max(i1, 0), DIM - 1);
}

// Trilinear sample of an interleaved vec3 field.
__device__ __forceinline__ void sample_vec3(const float* __restrict__ f,
                                            float cz, float cy, float cx,
                                            float out[3]) {
    int z0, z1, y0, y1, x0, x1;
    float wz0, wz1, wy0, wy1, wx0, wx1;
    axis_corners(cz, z0, z1, wz0, wz1);
    axis_corners(cy, y0, y1, wy0, wy1);
    axis_corners(cx, x0, x1, wx0, wx1);

    float a0 = 0.0f, a1 = 0.0f, a2 = 0.0f;
#pragma unroll
    for (int k = 0; k < 8; ++k) {
        int  zz = (k & 4) ? z1 : z0;
        int  yy = (k & 2) ? y1 : y0;
        int  xx = (k & 1) ? x1 : x0;
        float w = ((k & 4) ? wz1 : wz0) *
                  ((k & 2) ? wy1 : wy0) *
                  ((k & 1) ? wx1 : wx0);
        const float* p = f + 3 * ((zz << 14) + (yy << 7) + xx);
        a0 = fmaf(w, p[0], a0);
        a1 = fmaf(w, p[1], a1);
        a2 = fmaf(w, p[2], a2);
    }
    out[0] = a0; out[1] = a1; out[2] = a2;
}

// Trilinear sample of a scalar field.
__device__ __forceinline__ float sample_scalar(const float* __restrict__ f,
                                               float cz, float cy, float cx) {
    int z0, z1, y0, y1, x0, x1;
    float wz0, wz1, wy0, wy1, wx0, wx1;
    axis_corners(cz, z0, z1, wz0, wz1);
    axis_corners(cy, y0, y1, wy0, wy1);
    axis_corners(cx, x0, x1, wx0, wx1);

    float acc = 0.0f;
#pragma unroll
    for (int k = 0; k < 8; ++k) {
        int  zz = (k & 4) ? z1 : z0;
        int  yy = (k & 2) ? y1 : y0;
        int  xx = (k & 1) ? x1 : x0;
        float w = ((k & 4) ? wz1 : wz0) *
                  ((k & 2) ? wy1 : wy0) *
                  ((k & 1) ? wx1 : wx0);
        acc = fmaf(w, f[(zz << 14) + (yy << 7) + xx], acc);
    }
    return acc;
}

// Nearest sample; rintf == round-half-to-even, matching jnp.round.
__device__ __forceinline__ float sample_nearest(const float* __restrict__ f,
                                                float cz, float cy, float cx) {
    int iz = (int)rintf(cz);
    int iy = (int)rintf(cy);
    int ix = (int)rintf(cx);
    float m = (iz >= 0 && iz < DIM &&
               iy >= 0 && iy < DIM &&
               ix >= 0 && ix < DIM) ? 1.0f : 0.0f;
    int z = min(max(iz, 0), DIM - 1);
    int y = min(max(iy, 0), DIM - 1);
    int x = min(max(ix, 0), DIM - 1);
    return f[(z << 14) + (y << 7) + x] * m;
}

// ---------------------------------------------------------------------------
// Kernel 1: blend channel-major DVFs -> interleaved vec3, scaled by
// sign/2^NUM_STEPS. 4 voxels/thread, b128 reads per channel, b128 writes.
// ---------------------------------------------------------------------------
__global__ void __launch_bounds__(256)
blend_kernel(const float* __restrict__ dvf0, const float* __restrict__ dvf1,
             const float* __restrict__ w, float* __restrict__ out, float sign) {
    int i0 = (blockIdx.x * blockDim.x + threadIdx.x) * VPT;
    if (i0 >= NVOX) return;
    float wt = w[0];                     // broadcast scalar, L2-resident
    float s  = sign * (1.0f / (float)(1 << NUM_STEPS));

    float4 a[3], b[3];
#pragma unroll
    for (int c = 0; c < 3; ++c) {
        a[c] = *(const float4*)(dvf0 + (size_t)c * NVOX + i0);
        b[c] = *(const float4*)(dvf1 + (size_t)c * NVOX + i0);
    }
    float res[12];
#pragma unroll
    for (int c = 0; c < 3; ++c) {
        const float* ap = (const float*)&a[c];
        const float* bp = (const float*)&b[c];
#pragma unroll
        for (int j = 0; j < VPT; ++j)
            res[3 * j + c] = (ap[j] * (1.0f - wt) + bp[j] * wt) * s;
    }
    float4* po = (float4*)(out + (size_t)3 * i0);
    po[0] = make_float4(res[0], res[1], res[2],  res[3]);
    po[1] = make_float4(res[4], res[5], res[6],  res[7]);
    po[2] = make_float4(res[8], res[9], res[10], res[11]);
}

// ---------------------------------------------------------------------------
// Kernel 2: one scaling-and-squaring step: out = in + warp(in, in).
// 4 voxels/thread (all in the same x-row since 128 % 4 == 0), b128 streaming
// I/O, grid-stride with 2 iterations so global_prefetch_b8 covers the next
// iteration's streaming read while this iteration's 32 gathers (L2-resident:
// field is 24 MB << 192 MB L2) are in flight.
// ---------------------------------------------------------------------------
__global__ void __launch_bounds__(256)
compose_kernel(const float* __restrict__ in, float* __restrict__ out) {
    int stride = gridDim.x * blockDim.x * VPT;
    for (int i0 = (blockIdx.x * blockDim.x + threadIdx.x) * VPT; i0 < NVOX;
         i0 += stride) {
        // gfx1250 speculative prefetch (OOB translations silently dropped)
        __builtin_prefetch(in + (size_t)3 * (i0 + stride), 0, 0);

        const float4* pin = (const float4*)(in + (size_t)3 * i0);
        float4 qa = pin[0], qb = pin[1], qc = pin[2];
        float dd[12] = {qa.x, qa.y, qa.z, qa.w,
                        qb.x, qb.y, qb.z, qb.w,
                        qc.x, qc.y, qc.z, qc.w};
        float ro[12];

        int x0 = i0 & (DIM - 1);
        int y  = (i0 >> 7) & (DIM - 1);
        int z  = i0 >> 14;

#pragma unroll
        for (int j = 0; j < VPT; ++j) {
            float d0 = dd[3 * j], d1 = dd[3 * j + 1], d2 = dd[3 * j + 2];
            float s[3];
            sample_vec3(in, (float)z + d0, (float)y + d1,
                        (float)(x0 + j) + d2, s);
            ro[3 * j + 0] = d0 + s[0];
            ro[3 * j + 1] = d1 + s[1];
            ro[3 * j + 2] = d2 + s[2];
        }
        float4* po = (float4*)(out + (size_t)3 * i0);
        po[0] = make_float4(ro[0], ro[1], ro[2],  ro[3]);
        po[1] = make_float4(ro[4], ro[5], ro[6],  ro[7]);
        po[2] = make_float4(ro[8], ro[9], ro[10], ro[11]);
    }
}

// ---------------------------------------------------------------------------
// Kernel 3: FUSED final compose step + all three warps. The warp only needs
// the final forward ddf at the output voxel itself, so step 5 of the forward
// scaling-and-squaring is computed inline (never materialized): saves a full
// 24 MB write + 24 MB read pass. 4 voxels/thread, b128 output stores.
// ---------------------------------------------------------------------------
__global__ void __launch_bounds__(256)
fused_final_kernel(const float* __restrict__ ddf4,   // forward field, step 4
                   const float* __restrict__ image,
                   const float* __restrict__ cavity,
                   const float* __restrict__ coronary,
                   float* __restrict__ oimg,
                   float* __restrict__ ocav,
                   float* __restrict__ ocor) {
    int i0 = (blockIdx.x * blockDim.x + threadIdx.x) * VPT;
    if (i0 >= NVOX) return;

    const float4* pin = (const float4*)(ddf4 + (size_t)3 * i0);
    float4 qa = pin[0], qb = pin[1], qc = pin[2];
    float dd[12] = {qa.x, qa.y, qa.z, qa.w,
                    qb.x, qb.y, qb.z, qb.w,
                    qc.x, qc.y, qc.z, qc.w};

    int x0 = i0 & (DIM - 1);
    int y  = (i0 >> 7) & (DIM - 1);
    int z  = i0 >> 14;

    float rimg[VPT], rcav[VPT], rcor[VPT];
#pragma unroll
    for (int j = 0; j < VPT; ++j) {
        float d0 = dd[3 * j], d1 = dd[3 * j + 1], d2 = dd[3 * j + 2];
        // inline compose step 5: ddf5 = ddf4 + warp(ddf4, ddf4)  at this voxel
        float s[3];
        sample_vec3(ddf4, (float)z + d0, (float)y + d1,
                    (float)(x0 + j) + d2, s);
        float cz = (float)z        + d0 + s[0];
        float cy = (float)y        + d1 + s[1];
        float cx = (float)(x0 + j) + d2 + s[2];
        rimg[j] = sample_scalar(image, cz, cy, cx);
        rcav[j] = sample_nearest(cavity, cz, cy, cx);
        rcor[j] = sample_nearest(coronary, cz, cy, cx);
    }
    *(float4*)(oimg + i0) = make_float4(rimg[0], rimg[1], rimg[2], rimg[3]);
    *(float4*)(ocav + i0) = make_float4(rcav[0], rcav[1], rcav[2], rcav[3]);
    *(float4*)(ocor + i0) = make_float4(rcor[0], rcor[1], rcor[2], rcor[3]);
}

// ---------------------------------------------------------------------------
// Kernel 4: FUSED final inverse compose + point propagation. Instead of
// materializing inverse step 5 over all 2.1M voxels (~17M gathers + 48 MB),
// evaluate ddf5_inv(v) = ddf4(v) + warp(ddf4, ddf4)(v) on demand at the 8
// trilinear corner voxels of each of the 5000 points (~320K gathers total).
// Corner evaluation is skipped when its (mask-folded) weight is zero.
// ---------------------------------------------------------------------------
__global__ void __launch_bounds__(256)
fused_points_kernel(const float* __restrict__ f,    // inverse field, step 4
                    const float* __restrict__ pts,
                    float* __restrict__ opts, int npts) {
    int p = blockIdx.x * blockDim.x + threadIdx.x;
    if (p >= npts) return;
    float cz = pts[3 * p + 0];
    float cy = pts[3 * p + 1];
    float cx = pts[3 * p + 2];

    int z0, z1, y0, y1, x0, x1;
    float wz0, wz1, wy0, wy1, wx0, wx1;
    axis_corners(cz, z0, z1, wz0, wz1);
    axis_corners(cy, y0, y1, wy0, wy1);
    axis_corners(cx, x0, x1, wx0, wx1);

    float a0 = 0.0f, a1 = 0.0f, a2 = 0.0f;
#pragma unroll
    for (int k = 0; k < 8; ++k) {
        int  zz = (k & 4) ? z1 : z0;
        int  yy = (k & 2) ? y1 : y0;
        int  xx = (k & 1) ? x1 : x0;
        float w = ((k & 4) ? wz1 : wz0) *
                  ((k & 2) ? wy1 : wy0) *
                  ((k & 1) ? wx1 : wx0);
        if (w != 0.0f) {                 // each corner costs a full 8-tap compose
            const float* q = f + 3 * ((zz << 14) + (yy << 7) + xx);
            float d0 = q[0], d1 = q[1], d2 = q[2];
            float s[3];
            sample_vec3(f, (float)zz + d0, (float)yy + d1, (float)xx + d2, s);
            a0 = fmaf(w, d0 + s[0], a0);
            a1 = fmaf(w, d1 + s[1], a1);
            a2 = fmaf(w, d2 + s[2], a2);
        }
    }
    opts[3 * p + 0] = cz + a0;
    opts[3 * p + 1] = cy + a1;
    opts[3 * p + 2] = cx + a2;
}

// ---------------------------------------------------------------------------
// Host launcher. Graph-capture safe: only kernel launches on `stream`.
// Workspace: two interleaved vec3 field buffers (ping/pong), 48 MB total.
// ---------------------------------------------------------------------------
extern "C" void kernel_launch(void* const* d_in, const int* in_sizes, int n_in,
                              void* d_out, int out_size, void* d_ws, size_t ws_size,
                              hipStream_t stream) {
    const float* dvf0     = (const float*)d_in[0];   // [1,3,128,128,128]
    const float* dvf1     = (const float*)d_in[1];   // [1,3,128,128,128]
    const float* w        = (const float*)d_in[2];   // [1]
    const float* image    = (const float*)d_in[3];   // [1,1,128,128,128]
    const float* cavity   = (const float*)d_in[4];   // [1,1,128,128,128]
    const float* coronary = (const float*)d_in[5];   // [1,1,128,128,128]
    const float* points   = (const float*)d_in[6];   // [5000,3]
    const int npts = in_sizes[6] / 3;

    float* oimg = (float*)d_out;                     // N
    float* ocav = oimg + NVOX;                       // N
    float* ocor = ocav + NVOX;                       // N
    float* opts = ocor + NVOX;                       // npts*3

    float* bufA = (float*)d_ws;                      // 3*N floats
    float* bufB = bufA + (size_t)3 * NVOX;           // 3*N floats

    const int VEC_BLOCKS  = NVOX / (256 * VPT);      // 2048 (1 chunk/thread)
    const int COMP_BLOCKS = VEC_BLOCKS / 2;          // 1024 (2 iters -> prefetch)

    // --- forward field: ddf = dvf/32; 4 materialized compose steps ---
    blend_kernel<<<VEC_BLOCKS, 256, 0, stream>>>(dvf0, dvf1, w, bufA, 1.0f);
    const float* src = bufA;
    float*       dst = bufB;
    for (int s = 0; s < NUM_STEPS - 1; ++s) {        // steps 1..4: A->B->A->B->A
        compose_kernel<<<COMP_BLOCKS, 256, 0, stream>>>(src, dst);
        const float* t = src; src = dst; dst = (float*)t;
    }
    // step 5 fused with the three warps; forward result in `src` (== bufA)
    fused_final_kernel<<<VEC_BLOCKS, 256, 0, stream>>>(src, image, cavity,
                                                       coronary, oimg, ocav, ocor);

    // --- inverse field: ddf_inv = dvf2ddf(-dvf), reusing both buffers ---
    blend_kernel<<<VEC_BLOCKS, 256, 0, stream>>>(dvf0, dvf1, w, bufA, -1.0f);
    src = bufA;
    dst = bufB;
    for (int s = 0; s < NUM_STEPS - 1; ++s) {        // steps 1..4
        compose_kernel<<<COMP_BLOCKS, 256, 0, stream>>>(src, dst);
        const float* t = src; src = dst; dst = (float*)t;
    }
    // step 5 fused into the per-corner evaluation of the point kernel
    fused_points_kernel<<<(npts + 255) / 256, 256, 0, stream>>>(src, points,
                                                                opts, npts);
}